// BilinearFusion_66383014527553
// MI455X (gfx1250) — compile-verified
//
#include <hip/hip_runtime.h>

typedef float v2f __attribute__((ext_vector_type(2)));
typedef float v8f __attribute__((ext_vector_type(8)));

#define N_HEAD   8
#define DIM      128
#define ROWS_WG  128        // batch rows per workgroup
#define ROWS_WV  16         // batch rows per wave (8 waves/block)

// Block = (head, 128 rows). W3[head] staged once into LDS in WMMA-B operand
// order; 8 waves each run a 16-row WMMA GEMM against it.
__global__ __launch_bounds__(256) void bilinear_fusion_wmma(
    const float* __restrict__ x1, const float* __restrict__ x2,
    const float* __restrict__ W1, const float* __restrict__ b1,
    const float* __restrict__ W2, const float* __restrict__ W3,
    float* __restrict__ ws)                // [B][N_HEAD] partials
{
  // LDS: W3[h] repacked so that for (ntile, kk) lane L reads its B float2 at
  //   word ((ntile*32 + kk)*32 + L)*2      (conflict-free, immediate offsets)
  __shared__ float w3s[DIM * DIM];         // exactly 64 KB

  const int tid  = threadIdx.x;
  const int lane = tid & 31;
  const int wv   = tid >> 5;               // wave id -> row sub-tile
  const int h    = blockIdx.y;             // head
  const int half = lane >> 4;              // 0: lanes 0-15, 1: lanes 16-31
  const int l16  = lane & 15;
  const int waveRow = blockIdx.x * ROWS_WG + wv * ROWS_WV;

  // ---- cooperative stage of W3[h] into LDS, repacked to B-operand order ----
  {
    const float* g  = W3 + (size_t)h * DIM * DIM;
    const int   kk  = tid & 31;            // lane -> K group (coalesced reads)
    const int   nb  = tid >> 5;            // wave -> output row group
#pragma unroll
    for (int i = 0; i < 16; ++i) {
      const int np_ = i * 8 + nb;          // W3 output row o = n'
      float4 v = *(const float4*)(g + (size_t)np_ * DIM + kk * 4);
      const int ntile = np_ >> 4, l = np_ & 15;
      float* d = w3s + (size_t)((ntile * 32 + kk) * 32) * 2;
      v2f lo, hi;
      lo.x = v.x; lo.y = v.y;              // k = 4kk+0,+1  (lanes 0-15)
      hi.x = v.z; hi.y = v.w;              // k = 4kk+2,+3  (lanes 16-31)
      *(v2f*)(d + l * 2)        = lo;
      *(v2f*)(d + (l + 16) * 2) = hi;
    }
  }

  // ---- A tile: 16 rows x K=128 of head h from x1, in WMMA f32 A layout ----
  v2f a[32];
  {
    const float* ap = x1 + (size_t)(waveRow + l16) * 1024 + h * DIM + 2 * half;
#pragma unroll
    for (int kk = 0; kk < 32; ++kk)
      a[kk] = *(const v2f*)(ap + 4 * kk);
  }

  // ---- t1 = x1_h . W1_h + b1_h per row (tiny VALU dot, K split per half) ----
  float t1all;
  {
    const float* xp = x1 + (size_t)(waveRow + l16) * 1024 + h * DIM + 64 * half;
    const float* wp = W1 + h * DIM + 64 * half;
    float p = 0.f;
#pragma unroll
    for (int j = 0; j < 64; j += 4) {
      float4 xv = *(const float4*)(xp + j);
      float4 wv = *(const float4*)(wp + j);
      p += xv.x * wv.x + xv.y * wv.y + xv.z * wv.z + xv.w * wv.w;
    }
    p += __shfl_xor(p, 16);                // all lanes: full dot for row l16
    t1all = p + b1[h];
  }

  __syncthreads();                         // W3 staging visible

  // ---- bilinear: proj'[m,o] = sum_i x1[m,i]*W3[h,o,i] + W2[h,o] (C init) ----
  float s[8];
#pragma unroll
  for (int r = 0; r < 8; ++r) s[r] = 0.f;

  const float* bbase = w3s + lane * 2;

  for (int np = 0; np < 4; ++np) {         // two 16-col output tiles per iter
    const int n0 = 2 * np, n1 = 2 * np + 1;

    const float w20 = W2[h * DIM + n0 * 16 + l16];
    const float w21 = W2[h * DIM + n1 * 16 + l16];
    v8f c0, c1;
#pragma unroll
    for (int r = 0; r < 8; ++r) { c0[r] = w20; c1[r] = w21; }

#pragma unroll
    for (int kk = 0; kk < 32; ++kk) {
      v2f bb0 = *(const v2f*)(bbase + (n0 * 32 + kk) * 64);   // ds_load_b64
      v2f bb1 = *(const v2f*)(bbase + (n1 * 32 + kk) * 64);
      c0 = __builtin_amdgcn_wmma_f32_16x16x4_f32(false, a[kk], false, bb0,
                                                 (short)0, c0, false, false);
      c1 = __builtin_amdgcn_wmma_f32_16x16x4_f32(false, a[kk], false, bb1,
                                                 (short)0, c1, false, false);
    }

    // elementwise multiply with the matching x2 tile, accumulate row partials
#pragma unroll
    for (int r = 0; r < 8; ++r) {
      const float* x2p = x2 + (size_t)(waveRow + r + 8 * half) * 1024 + h * DIM;
      const float m0 = x2p[n0 * 16 + l16];
      const float m1 = x2p[n1 * 16 + l16];
      s[r] += c0[r] * m0 + c1[r] * m1;
    }
  }

  // ---- per-row reduce over N (16 lanes of each half), fold in t1, store ----
#pragma unroll
  for (int r = 0; r < 8; ++r) {
    float v = s[r];
    v += __shfl_xor(v, 1);
    v += __shfl_xor(v, 2);
    v += __shfl_xor(v, 4);
    v += __shfl_xor(v, 8);
    // t1 for this row lives on lane (r + 8*half); pull it to lanes 0/16
    const float t = __shfl(t1all, r + 8 * half);
    if (l16 == 0)
      ws[(size_t)(waveRow + r + 8 * half) * N_HEAD + h] = v + t;
  }
}

// out[i] = sum over heads of ws[i][h]  (deterministic, no atomics)
__global__ __launch_bounds__(256) void reduce_heads(
    const float* __restrict__ ws, float* __restrict__ out, int n)
{
  const int i = blockIdx.x * 256 + threadIdx.x;
  if (i < n) {
    const float* p = ws + (size_t)i * N_HEAD;
    float acc = 0.f;
#pragma unroll
    for (int hh = 0; hh < N_HEAD; ++hh) acc += p[hh];
    out[i] = acc;
  }
}

extern "C" void kernel_launch(void* const* d_in, const int* in_sizes, int n_in,
                              void* d_out, int out_size, void* d_ws, size_t ws_size,
                              hipStream_t stream) {
  (void)n_in; (void)ws_size; (void)in_sizes;
  const float* x1 = (const float*)d_in[0];
  const float* x2 = (const float*)d_in[1];
  const float* W1 = (const float*)d_in[2];
  const float* b1 = (const float*)d_in[3];
  const float* W2 = (const float*)d_in[4];
  const float* W3 = (const float*)d_in[5];
  float* out = (float*)d_out;
  float* ws  = (float*)d_ws;               // needs B * N_HEAD * 4 = 512 KB

  const int Bn = out_size;                 // 16384
  dim3 grid(Bn / ROWS_WG, N_HEAD);
  bilinear_fusion_wmma<<<grid, 256, 0, stream>>>(x1, x2, W1, b1, W2, W3, ws);
  reduce_heads<<<(Bn + 255) / 256, 256, 0, stream>>>(ws, out, Bn);
}